// dgl_Homo_MLPLinkPredictionPredictor_39625368273429
// MI455X (gfx1250) — compile-verified
//
#include <hip/hip_runtime.h>

typedef __attribute__((ext_vector_type(2))) float v2f;
typedef __attribute__((ext_vector_type(8))) float v8f;

#define TILES_PER_BLOCK 8
#define BLOCK_THREADS   256

// One wave32 per 16-edge tile.
// Layer 1: [16x128] (gathered, fp32) @ [128x64] via V_WMMA_F32_16X16X4_F32,
//          K split into 32 steps of 4; 4 N-tiles of 16 -> 128 WMMAs/tile.
// Layer 2: [16x64] @ [64x2] via per-lane FMAs + 16-lane shfl_xor reduction.
__global__ __launch_bounds__(BLOCK_THREADS)
void edge_mlp_wmma_kernel(const float* __restrict__ h,
                          const int*   __restrict__ src,
                          const int*   __restrict__ dst,
                          const float* __restrict__ W1,
                          const float* __restrict__ B1,
                          const float* __restrict__ W2,
                          const float* __restrict__ B2,
                          float* __restrict__ out,
                          int nEdges)
{
    // W1 pre-swizzled into per-(kstep,ntile,lane) B-fragments: 4096 float2 = 32KB
    __shared__ v2f ldsW1[4096];

    const int tid = threadIdx.x;

    // Stage + swizzle W1 -> LDS. Fragment layout (ISA 7.12.2, 32-bit B 4x16):
    // lane<16: {B[k0][n], B[k0+1][n]}, lane>=16: {B[k0+2][n], B[k0+3][n]}.
    for (int i = tid; i < 4096; i += BLOCK_THREADS) {
        int L  = i & 31;
        int t  = (i >> 5) & 3;   // N-tile 0..3
        int s  = i >> 7;         // K-step 0..31
        int hi = L >> 4;
        int lo = L & 15;
        int k0 = 4 * s + 2 * hi;
        int n  = 16 * t + lo;
        v2f frag;
        frag.x = W1[(k0 + 0) * 64 + n];
        frag.y = W1[(k0 + 1) * 64 + n];
        ldsW1[i] = frag;
    }
    __syncthreads();

    const int wave = tid >> 5;
    const int lane = tid & 31;
    const int hi   = lane >> 4;   // selects K pair within fragment / M high bit in C
    const int lo   = lane & 15;   // row within tile (for A) / N within ntile (for B,C)

    const int nTiles = (nEdges + 15) >> 4;
    const int tile   = blockIdx.x * TILES_PER_BLOCK + wave;
    if (tile >= nTiles) return;

    // Per-lane gathered row pointers (lanes 16..31 mirror 0..15's rows).
    const int eRow = tile * 16 + lo;
    const int eCl  = eRow < nEdges ? eRow : (nEdges - 1);
    const float* __restrict__ pS = h + (long long)src[eCl] * 64;
    const float* __restrict__ pD = h + (long long)dst[eCl] * 64;

    v8f acc0 = {}, acc1 = {}, acc2 = {}, acc3 = {};

    // K-steps 0..15: columns from h[src] (he[:, 0:64])
#pragma unroll
    for (int s = 0; s < 16; ++s) {
        v2f a = *(const v2f*)(pS + 4 * s + 2 * hi);
        const v2f* f = &ldsW1[(s * 4) * 32 + lane];
        v2f b0 = f[0 * 32], b1 = f[1 * 32], b2 = f[2 * 32], b3 = f[3 * 32];
        acc0 = __builtin_amdgcn_wmma_f32_16x16x4_f32(false, a, false, b0, (short)0, acc0, false, false);
        acc1 = __builtin_amdgcn_wmma_f32_16x16x4_f32(false, a, false, b1, (short)0, acc1, false, false);
        acc2 = __builtin_amdgcn_wmma_f32_16x16x4_f32(false, a, false, b2, (short)0, acc2, false, false);
        acc3 = __builtin_amdgcn_wmma_f32_16x16x4_f32(false, a, false, b3, (short)0, acc3, false, false);
    }
    // K-steps 16..31: columns from h[dst] (he[:, 64:128])
#pragma unroll
    for (int s = 16; s < 32; ++s) {
        v2f a = *(const v2f*)(pD + 4 * (s - 16) + 2 * hi);
        const v2f* f = &ldsW1[(s * 4) * 32 + lane];
        v2f b0 = f[0 * 32], b1 = f[1 * 32], b2 = f[2 * 32], b3 = f[3 * 32];
        acc0 = __builtin_amdgcn_wmma_f32_16x16x4_f32(false, a, false, b0, (short)0, acc0, false, false);
        acc1 = __builtin_amdgcn_wmma_f32_16x16x4_f32(false, a, false, b1, (short)0, acc1, false, false);
        acc2 = __builtin_amdgcn_wmma_f32_16x16x4_f32(false, a, false, b2, (short)0, acc2, false, false);
        acc3 = __builtin_amdgcn_wmma_f32_16x16x4_f32(false, a, false, b3, (short)0, acc3, false, false);
    }

    // Layer-1 bias + ReLU, then layer-2 per-lane partial dot products.
    // C/D layout: VGPR j, lane L -> M = j + 8*hi, N = (L&15) (+16*t).
    float bias0 = B1[ 0 + lo];
    float bias1 = B1[16 + lo];
    float bias2 = B1[32 + lo];
    float bias3 = B1[48 + lo];
    float w2a0 = W2[( 0 + lo) * 2 + 0], w2b0 = W2[( 0 + lo) * 2 + 1];
    float w2a1 = W2[(16 + lo) * 2 + 0], w2b1 = W2[(16 + lo) * 2 + 1];
    float w2a2 = W2[(32 + lo) * 2 + 0], w2b2 = W2[(32 + lo) * 2 + 1];
    float w2a3 = W2[(48 + lo) * 2 + 0], w2b3 = W2[(48 + lo) * 2 + 1];

    float r0[8], r1[8];
#pragma unroll
    for (int j = 0; j < 8; ++j) {
        float h0 = fmaxf(acc0[j] + bias0, 0.0f);
        float h1 = fmaxf(acc1[j] + bias1, 0.0f);
        float h2 = fmaxf(acc2[j] + bias2, 0.0f);
        float h3 = fmaxf(acc3[j] + bias3, 0.0f);
        r0[j] = h0 * w2a0 + h1 * w2a1 + h2 * w2a2 + h3 * w2a3;
        r1[j] = h0 * w2b0 + h1 * w2b1 + h2 * w2b2 + h3 * w2b3;
    }

    // Reduce over the 16 N-lanes (xor masks 1..8 never cross the half-wave).
#pragma unroll
    for (int m = 1; m < 16; m <<= 1) {
#pragma unroll
        for (int j = 0; j < 8; ++j) {
            r0[j] += __shfl_xor(r0[j], m, 32);
            r1[j] += __shfl_xor(r1[j], m, 32);
        }
    }

    if (lo < 2) {
        const float bb = (lo == 0) ? B2[0] : B2[1];
#pragma unroll
        for (int j = 0; j < 8; ++j) {
            int m = j + 8 * hi;
            int e = tile * 16 + m;
            if (e < nEdges) {
                float v = ((lo == 0) ? r0[j] : r1[j]) + bb;
                out[e * 2 + lo] = v;
            }
        }
    }
}

extern "C" void kernel_launch(void* const* d_in, const int* in_sizes, int n_in,
                              void* d_out, int out_size, void* d_ws, size_t ws_size,
                              hipStream_t stream) {
    const float* h   = (const float*)d_in[0];
    const int*   src = (const int*)d_in[1];
    const int*   dst = (const int*)d_in[2];
    const float* W1  = (const float*)d_in[3];
    const float* B1  = (const float*)d_in[4];
    const float* W2  = (const float*)d_in[5];
    const float* B2  = (const float*)d_in[6];
    float* out = (float*)d_out;

    const int nEdges = in_sizes[1];
    const int nTiles = (nEdges + 15) >> 4;
    const int nBlocks = (nTiles + TILES_PER_BLOCK - 1) / TILES_PER_BLOCK;

    edge_mlp_wmma_kernel<<<nBlocks, BLOCK_THREADS, 0, stream>>>(
        h, src, dst, W1, B1, W2, B2, out, nEdges);
}